// GCN_68882685493292
// MI455X (gfx1250) — compile-verified
//
#include <hip/hip_runtime.h>
#include <math.h>

#define N_NODES 100000
#define NUM_E   1600000
#define IN_C    128
#define HID     256
#define BN_EPS_F 1e-5f
#define L2_EPS_F 1e-12f

typedef __attribute__((ext_vector_type(2))) float v2f;
typedef __attribute__((ext_vector_type(8))) float v8f;

// ---------------------------------------------------------------------------
// GEMM: Y[n, c] = sum_k X[n, k] * W[c, k] (+ bias[c])   (i.e. Y = X @ W^T + b)
// Block: 256 threads = 8 waves. Block covers 16 rows x 128 cols.
// Each wave computes one 16x16 tile with V_WMMA_F32_16X16X4_F32.
// ---------------------------------------------------------------------------
template <int K>
__global__ __launch_bounds__(256) void gemm_wmma(
    const float* __restrict__ X, const float* __restrict__ W,
    const float* __restrict__ bias, float* __restrict__ Y, int Hout) {
  constexpr int XS_STRIDE = K + 4;  // pad: stride % 64 banks == 4 -> no conflicts
  __shared__ float Xs[16 * XS_STRIDE];

  const int rowBase = blockIdx.x * 16;

  // Cooperative stage of the 16 x K X-tile into LDS.
  for (int idx = threadIdx.x; idx < 16 * K; idx += 256) {
    const int r = idx / K;
    const int c = idx - r * K;
    Xs[r * XS_STRIDE + c] = X[(size_t)(rowBase + r) * K + c];
  }
  __syncthreads();

  const int lane = threadIdx.x & 31;
  const int wave = threadIdx.x >> 5;
  const int colBase = (blockIdx.y * 8 + wave) * 16;
  const int mn = lane & 15;             // M index for A, N index for B/C
  const int khi = (lane >> 4) << 1;     // lanes 0-15 -> K {0,1}; lanes 16-31 -> K {2,3}

  const float* __restrict__ Wrow = W + (size_t)(colBase + mn) * K;
  const float* __restrict__ Xrow = Xs + mn * XS_STRIDE;

  v8f acc = {0.f, 0.f, 0.f, 0.f, 0.f, 0.f, 0.f, 0.f};

#pragma unroll
  for (int k0 = 0; k0 < K; k0 += 4) {
    v2f a, b;
    a.x = Xrow[k0 + khi];
    a.y = Xrow[k0 + khi + 1];
    b.x = Wrow[k0 + khi];
    b.y = Wrow[k0 + khi + 1];
    // 8 args: (neg_a, A, neg_b, B, c_mod, C, reuse_a, reuse_b)
    acc = __builtin_amdgcn_wmma_f32_16x16x4_f32(
        false, a, false, b, (short)0, acc, false, false);
  }

  const float bb = bias ? bias[colBase + mn] : 0.0f;
  const int rOff = (lane >> 4) << 3;  // upper half-wave holds M = i + 8
#pragma unroll
  for (int i = 0; i < 8; ++i) {
    Y[(size_t)(rowBase + rOff + i) * Hout + colBase + mn] = acc[i] + bb;
  }
}

// ---------------------------------------------------------------------------
// Degree / normalization
// ---------------------------------------------------------------------------
__global__ void zero_deg_kernel(float* __restrict__ deg) {
  const int n = blockIdx.x * 256 + threadIdx.x;
  if (n < N_NODES) deg[n] = 0.0f;
}

__global__ void deg_accum_kernel(const int* __restrict__ ei,
                                 const float* __restrict__ ew,
                                 float* __restrict__ deg) {
  const int e = blockIdx.x * 256 + threadIdx.x;
  if (e < NUM_E) atomicAdd(&deg[ei[NUM_E + e]], ew[e]);  // dst row
}

__global__ void dinv_kernel(float* __restrict__ deg) {
  const int n = blockIdx.x * 256 + threadIdx.x;
  if (n < N_NODES) deg[n] = rsqrtf(deg[n] + 1.0f);  // deg+1 >= 1 always
}

// agg[n,h] = bias[h] + dinv[n]^2 * h[n,h]   (self-loop term + conv bias)
__global__ void init_agg_kernel(const float* __restrict__ Hm,
                                const float* __restrict__ bias,
                                const float* __restrict__ dinv,
                                float* __restrict__ Agg) {
  const size_t i = (size_t)blockIdx.x * 256 + threadIdx.x;
  if (i >= (size_t)N_NODES * HID) return;
  const int n = (int)(i >> 8);
  const int h = (int)(i & 255);
  const float di = dinv[n];
  Agg[i] = bias[h] + di * di * Hm[i];
}

// One wave per edge: agg[dst,:] += dinv[src]*ew*dinv[dst] * h[src,:]
__global__ __launch_bounds__(256) void edge_agg_kernel(
    const int* __restrict__ ei, const float* __restrict__ ew,
    const float* __restrict__ dinv, const float* __restrict__ Hm,
    float* __restrict__ Agg) {
  const int gw = (blockIdx.x * 256 + threadIdx.x) >> 5;
  const int lane = threadIdx.x & 31;
  if (gw >= NUM_E) return;
  const int s = ei[gw];
  const int d = ei[NUM_E + gw];
  const float coef = dinv[s] * ew[gw] * dinv[d];
  const float* __restrict__ hs = Hm + (size_t)s * HID;
  float* __restrict__ ag = Agg + (size_t)d * HID;
#pragma unroll
  for (int i = 0; i < 8; ++i) {
    const int h = lane + (i << 5);
    atomicAdd(ag + h, coef * hs[h]);
  }
}

// ---------------------------------------------------------------------------
// Post layer 1: x1 = l2norm(relu(bn(agg))) + 0.2*x0     (one wave per node)
// ---------------------------------------------------------------------------
__global__ __launch_bounds__(256) void post1_kernel(
    const float* __restrict__ Agg, const float* __restrict__ g,
    const float* __restrict__ be, const float* __restrict__ m,
    const float* __restrict__ v, const float* __restrict__ x0,
    float* __restrict__ x1) {
  const int node = blockIdx.x * 8 + (threadIdx.x >> 5);
  const int lane = threadIdx.x & 31;
  if (node >= N_NODES) return;
  const float* __restrict__ a = Agg + (size_t)node * HID;
  float vals[8];
  float ss = 0.0f;
#pragma unroll
  for (int i = 0; i < 8; ++i) {
    const int h = lane + (i << 5);
    float t = (a[h] - m[h]) * rsqrtf(v[h] + BN_EPS_F) * g[h] + be[h];
    t = fmaxf(t, 0.0f);
    vals[i] = t;
    ss += t * t;
  }
#pragma unroll
  for (int o = 16; o > 0; o >>= 1) ss += __shfl_xor(ss, o, 32);
  const float inv = 1.0f / fmaxf(sqrtf(ss), L2_EPS_F);
  const float* __restrict__ x0r = x0 + (size_t)node * HID;
  float* __restrict__ x1r = x1 + (size_t)node * HID;
#pragma unroll
  for (int i = 0; i < 8; ++i) {
    const int h = lane + (i << 5);
    x1r[h] = vals[i] * inv + 0.2f * x0r[h];
  }
}

// ---------------------------------------------------------------------------
// Post layer 2: emb = l2norm(bn(agg)) + 0.5*x1 ; logits = emb @ Wc^T + bc
// ---------------------------------------------------------------------------
__global__ __launch_bounds__(256) void post2_kernel(
    const float* __restrict__ Agg, const float* __restrict__ g,
    const float* __restrict__ be, const float* __restrict__ m,
    const float* __restrict__ v, const float* __restrict__ x1,
    const float* __restrict__ Wc, const float* __restrict__ bc,
    float* __restrict__ emb, float* __restrict__ logits) {
  const int node = blockIdx.x * 8 + (threadIdx.x >> 5);
  const int lane = threadIdx.x & 31;
  if (node >= N_NODES) return;
  const float* __restrict__ a = Agg + (size_t)node * HID;
  float vals[8];
  float ss = 0.0f;
#pragma unroll
  for (int i = 0; i < 8; ++i) {
    const int h = lane + (i << 5);
    const float t = (a[h] - m[h]) * rsqrtf(v[h] + BN_EPS_F) * g[h] + be[h];
    vals[i] = t;
    ss += t * t;
  }
#pragma unroll
  for (int o = 16; o > 0; o >>= 1) ss += __shfl_xor(ss, o, 32);
  const float inv = 1.0f / fmaxf(sqrtf(ss), L2_EPS_F);
  const float* __restrict__ x1r = x1 + (size_t)node * HID;
  float* __restrict__ er = emb + (size_t)node * HID;
  float l0 = 0.0f, l1 = 0.0f;
#pragma unroll
  for (int i = 0; i < 8; ++i) {
    const int h = lane + (i << 5);
    const float e = vals[i] * inv + 0.5f * x1r[h];
    er[h] = e;
    l0 += e * Wc[h];
    l1 += e * Wc[HID + h];
  }
#pragma unroll
  for (int o = 16; o > 0; o >>= 1) {
    l0 += __shfl_xor(l0, o, 32);
    l1 += __shfl_xor(l1, o, 32);
  }
  if (lane == 0) {
    logits[(size_t)node * 2 + 0] = l0 + bc[0];
    logits[(size_t)node * 2 + 1] = l1 + bc[1];
  }
}

// ---------------------------------------------------------------------------
extern "C" void kernel_launch(void* const* d_in, const int* in_sizes, int n_in,
                              void* d_out, int out_size, void* d_ws,
                              size_t ws_size, hipStream_t stream) {
  const float* x   = (const float*)d_in[0];
  const int*   ei  = (const int*)d_in[1];   // [2,E]: row0=src, row1=dst
  const float* ew  = (const float*)d_in[2];
  const float* W0  = (const float*)d_in[3];
  const float* b0  = (const float*)d_in[4];
  const float* W1  = (const float*)d_in[5];
  const float* b1  = (const float*)d_in[6];
  const float* g1  = (const float*)d_in[7];
  const float* be1 = (const float*)d_in[8];
  const float* m1  = (const float*)d_in[9];
  const float* v1  = (const float*)d_in[10];
  const float* W2  = (const float*)d_in[11];
  const float* b2  = (const float*)d_in[12];
  const float* g2  = (const float*)d_in[13];
  const float* be2 = (const float*)d_in[14];
  const float* m2  = (const float*)d_in[15];
  const float* v2  = (const float*)d_in[16];
  const float* Wc  = (const float*)d_in[17];
  const float* bc  = (const float*)d_in[18];

  float* out    = (float*)d_out;
  float* emb    = out;                              // [N, 256]
  float* logits = out + (size_t)N_NODES * HID;      // [N, 2]

  const size_t NH = (size_t)N_NODES * HID;
  float* ws   = (float*)d_ws;
  float* x0   = ws;             // [N,H] lin0 skip
  float* hbuf = ws + NH;        // [N,H] h1 then h2
  float* agg  = ws + 2 * NH;    // [N,H] aggregation
  float* x1   = ws + 3 * NH;    // [N,H] layer-1 output
  float* dinv = ws + 4 * NH;    // [N]   deg -> dinv (in place)

  const dim3 blk(256);
  const dim3 gemm_grid(N_NODES / 16, HID / 128);  // 6250 x 2
  const int nodeBlocks = (N_NODES + 255) / 256;
  const int edgeBlocks = (NUM_E + 255) / 256;
  const int nhBlocks   = (int)((NH + 255) / 256);
  const int waveEdgeBlocks = (NUM_E * 32 + 255) / 256;  // 1 wave per edge
  const int nodeWaveBlocks = (N_NODES + 7) / 8;         // 1 wave per node

  // Degrees -> dinv (shared by both layers)
  zero_deg_kernel<<<nodeBlocks, blk, 0, stream>>>(dinv);
  deg_accum_kernel<<<edgeBlocks, blk, 0, stream>>>(ei, ew, dinv);
  dinv_kernel<<<nodeBlocks, blk, 0, stream>>>(dinv);

  // x0 = x @ W0^T + b0 ; h1 = x @ W1^T
  gemm_wmma<IN_C><<<gemm_grid, blk, 0, stream>>>(x, W0, b0, x0, HID);
  gemm_wmma<IN_C><<<gemm_grid, blk, 0, stream>>>(x, W1, nullptr, hbuf, HID);

  // Layer 1 aggregation + post
  init_agg_kernel<<<nhBlocks, blk, 0, stream>>>(hbuf, b1, dinv, agg);
  edge_agg_kernel<<<waveEdgeBlocks, blk, 0, stream>>>(ei, ew, dinv, hbuf, agg);
  post1_kernel<<<nodeWaveBlocks, blk, 0, stream>>>(agg, g1, be1, m1, v1, x0, x1);

  // h2 = x1 @ W2^T
  gemm_wmma<HID><<<gemm_grid, blk, 0, stream>>>(x1, W2, nullptr, hbuf, HID);

  // Layer 2 aggregation + post (+ fused classifier)
  init_agg_kernel<<<nhBlocks, blk, 0, stream>>>(hbuf, b2, dinv, agg);
  edge_agg_kernel<<<waveEdgeBlocks, blk, 0, stream>>>(ei, ew, dinv, hbuf, agg);
  post2_kernel<<<nodeWaveBlocks, blk, 0, stream>>>(agg, g2, be2, m2, v2, x1, Wc,
                                                   bc, emb, logits);
}